// GeoSpecNetLoss_20409684590742
// MI455X (gfx1250) — compile-verified
//
#include <hip/hip_runtime.h>
#include <hip/hip_bf16.h>
#include <math.h>

typedef float v2f __attribute__((ext_vector_type(2)));
typedef float v8f __attribute__((ext_vector_type(8)));

#define BATCH    8
#define N_COARSE 512
#define N_FINE1  2048
#define N_FINE2  4096
#define N_GT     4096
#define N_PART   2048
#define KNN      10

#define RTILES   4              // row tiles (of 16) per wave -> 64 rows/block

// workspace layout (float offsets)
#define OFF_ROW_C   0
#define OFF_ROW_F1  (OFF_ROW_C  + BATCH*N_COARSE)
#define OFF_ROW_F2  (OFF_ROW_F1 + BATCH*N_FINE1)
#define OFF_ROW_P   (OFF_ROW_F2 + BATCH*N_FINE2)
#define OFF_COL_C   (OFF_ROW_P  + BATCH*N_PART)
#define OFF_COL_F1  (OFF_COL_C  + BATCH*N_GT)
#define OFF_COL_F2  (OFF_COL_F1 + BATCH*N_GT)
#define OFF_SMOOTH  (OFF_COL_F2 + BATCH*N_GT)

// -------------------------------------------------------------------------
// init: column-min buffers must start at +inf every call (graph replay safe)
// -------------------------------------------------------------------------
__global__ void init_colmin_kernel(float* __restrict__ ws) {
    int i = blockIdx.x * blockDim.x + threadIdx.x;
    int n = 3 * BATCH * N_GT;
    if (i < n) ws[OFF_COL_C + i] = __builtin_inff();
}

// -------------------------------------------------------------------------
// Pairwise squared distances via V_WMMA_F32_16X16X4_F32.
//   d(m,n) = |a_m|^2 + |b_n|^2 - 2 a_m . b_n   (K=3 padded to K=4 with 0)
// One wave per (batch, 64-row group): 4 resident A-tiles, 4 WMMAs per
// loaded B-tile. B-tile loads are double-buffered in registers; the tile
// after next is prefetched. Row mins are register-resident and reduced with
// in-half shfl_xor; column mins merge via uint atomicMin (distances >= 0).
// -------------------------------------------------------------------------
template <bool WRITE_COL>
__global__ __launch_bounds__(32)
void pairdist_min_kernel(const float* __restrict__ A, const float* __restrict__ Bm,
                         float* __restrict__ rowmin, unsigned int* __restrict__ colminBits,
                         int N, int M) {
    const int b    = blockIdx.y;
    const int lane = threadIdx.x;
    const int hi   = lane >> 4;      // 0 -> K0/K1 (m 0..7), 1 -> K2/K3 (m 8..15)
    const int l    = lane & 15;

    const float* Ab = A  + (size_t)b * N * 3;
    const float* Bb = Bm + (size_t)b * M * 3;

    // ---- load 4 A-tiles (rows base .. base+63) --------------------------
    const int rowBase = blockIdx.x * (16 * RTILES);
    v2f aT[RTILES];
    __shared__ float a2s[16 * RTILES];
#pragma unroll
    for (int t = 0; t < RTILES; ++t) {
        const int row = rowBase + t * 16 + l;
        const float ax = Ab[row * 3 + 0];
        const float ay = Ab[row * 3 + 1];
        const float az = Ab[row * 3 + 2];
        aT[t].x = hi ? az : ax;
        aT[t].y = hi ? 0.0f : ay;
        if (hi == 0) a2s[t * 16 + l] = ax * ax + ay * ay + az * az;
    }
    __syncthreads();
    float a2r[RTILES][8];
#pragma unroll
    for (int t = 0; t < RTILES; ++t)
#pragma unroll
        for (int r = 0; r < 8; ++r) a2r[t][r] = a2s[t * 16 + r + 8 * hi];  // m = r + 8*hi

    float rmin[RTILES][8];
#pragma unroll
    for (int t = 0; t < RTILES; ++t)
#pragma unroll
        for (int r = 0; r < 8; ++r) rmin[t][r] = __builtin_inff();

    // ---- column-tile loop, register double-buffered ---------------------
    const int ntiles = M >> 4;
    float bx = Bb[l * 3 + 0];
    float by = Bb[l * 3 + 1];
    float bz = Bb[l * 3 + 2];

    for (int ct = 0; ct < ntiles; ++ct) {
        // in-flight load of next tile (branchless: clamp to last valid tile)
        const int ctn = (ct + 1 < ntiles) ? (ct + 1) : ct;
        const float nbx = Bb[(ctn * 16 + l) * 3 + 0];
        const float nby = Bb[(ctn * 16 + l) * 3 + 1];
        const float nbz = Bb[(ctn * 16 + l) * 3 + 2];
        if (ct + 2 < ntiles) __builtin_prefetch(&Bb[((ct + 2) * 16 + l) * 3], 0, 3);

        // B-tile (4x16 f32, 2 VGPRs): mirrors A-tile with N in place of M
        v2f bT;
        bT.x = hi ? bz : bx;
        bT.y = hi ? 0.0f : by;
        const float b2 = bx * bx + by * by + bz * bz;  // this lane's column norm

        float cmin = __builtin_inff();
#pragma unroll
        for (int t = 0; t < RTILES; ++t) {
            v8f acc = {};
            acc = __builtin_amdgcn_wmma_f32_16x16x4_f32(
                /*neg_a=*/false, aT[t], /*neg_b=*/false, bT,
                /*c_mod=*/(short)0, acc, /*reuse_a=*/false, /*reuse_b=*/false);
#pragma unroll
            for (int r = 0; r < 8; ++r) {
                float d = fmaxf(a2r[t][r] + b2 - 2.0f * acc[r], 0.0f);
                rmin[t][r] = fminf(rmin[t][r], d);
                cmin       = fminf(cmin, d);
            }
        }
        if (WRITE_COL) {
            cmin = fminf(cmin, __shfl_xor(cmin, 16));   // combine m 0..7 with m 8..15
            if (hi == 0)
                atomicMin(&colminBits[(size_t)b * M + ct * 16 + l], __float_as_uint(cmin));
        }
        bx = nbx; by = nby; bz = nbz;
    }

    // ---- row-min: reduce across the 16 lanes of each half ---------------
#pragma unroll
    for (int t = 0; t < RTILES; ++t) {
#pragma unroll
        for (int r = 0; r < 8; ++r) {
            float v = rmin[t][r];
            v = fminf(v, __shfl_xor(v, 1));
            v = fminf(v, __shfl_xor(v, 2));
            v = fminf(v, __shfl_xor(v, 4));
            v = fminf(v, __shfl_xor(v, 8));
            rmin[t][r] = v;
        }
    }
    if (l == 0) {
#pragma unroll
        for (int t = 0; t < RTILES; ++t)
#pragma unroll
            for (int r = 0; r < 8; ++r)
                rowmin[(size_t)b * N + rowBase + t * 16 + 8 * hi + r] = rmin[t][r];
    }
}

// -------------------------------------------------------------------------
// Smoothness: for each point of fine2, sum of squared distances to its 10
// nearest neighbors (self distance 0 occupies slot 0 and is excluded).
// -------------------------------------------------------------------------
__global__ __launch_bounds__(128)
void smooth_kernel(const float* __restrict__ pts, float* __restrict__ out) {
    const int b = blockIdx.y;
    const int i = blockIdx.x * blockDim.x + threadIdx.x;
    const float* P = pts + (size_t)b * N_FINE2 * 3;
    const float px = P[i * 3 + 0];
    const float py = P[i * 3 + 1];
    const float pz = P[i * 3 + 2];

    float best[KNN + 1];
#pragma unroll
    for (int t = 0; t <= KNN; ++t) best[t] = __builtin_inff();

    for (int j = 0; j < N_FINE2; ++j) {
        const float dx = P[j * 3 + 0] - px;
        const float dy = P[j * 3 + 1] - py;
        const float dz = P[j * 3 + 2] - pz;
        const float d = dx * dx + dy * dy + dz * dz;
        if (d < best[KNN]) {
            best[KNN] = d;
#pragma unroll
            for (int t = KNN; t > 0; --t) {    // one bubble pass keeps list sorted
                const float lo  = fminf(best[t - 1], best[t]);
                const float hiv = fmaxf(best[t - 1], best[t]);
                best[t - 1] = lo;
                best[t] = hiv;
            }
        }
    }
    float s = 0.0f;
#pragma unroll
    for (int t = 1; t <= KNN; ++t) s += best[t];   // skip self (best[0] == 0)
    out[(size_t)b * N_FINE2 + i] = s;
}

// -------------------------------------------------------------------------
// Final reduction: 1 block, 256 threads -> 6 scalars.
// -------------------------------------------------------------------------
__device__ float blockSum(float v, float* sdata) {
    const int t = threadIdx.x;
    sdata[t] = v;
    __syncthreads();
    for (int s = 128; s > 0; s >>= 1) {
        if (t < s) sdata[t] += sdata[t + s];
        __syncthreads();
    }
    const float r = sdata[0];
    __syncthreads();
    return r;
}

__global__ __launch_bounds__(256)
void reduce_kernel(const float* __restrict__ ws, float* __restrict__ out) {
    __shared__ float sdata[256];
    const int t = threadIdx.x;

    auto sumRegion = [&](int off, int n, bool dosqrt) -> float {
        float s = 0.0f;
        for (int i = t; i < n; i += 256) {
            const float v = ws[off + i];
            s += dosqrt ? sqrtf(v) : v;
        }
        return blockSum(s, sdata);
    };

    const float sRowC  = sumRegion(OFF_ROW_C,  BATCH * N_COARSE, false);
    const float sRowF1 = sumRegion(OFF_ROW_F1, BATCH * N_FINE1,  false);
    const float sRowF2 = sumRegion(OFF_ROW_F2, BATCH * N_FINE2,  false);
    const float sPart  = sumRegion(OFF_ROW_P,  BATCH * N_PART,   true);
    const float sColC  = sumRegion(OFF_COL_C,  BATCH * N_GT,     false);
    const float sColF1 = sumRegion(OFF_COL_F1, BATCH * N_GT,     false);
    const float sColF2 = sumRegion(OFF_COL_F2, BATCH * N_GT,     false);
    const float sSm    = sumRegion(OFF_SMOOTH, BATCH * N_FINE2,  false);

    if (t == 0) {
        const float cdC  = sRowC  / (float)(BATCH * N_COARSE) + sColC  / (float)(BATCH * N_GT);
        const float cdF1 = sRowF1 / (float)(BATCH * N_FINE1)  + sColF1 / (float)(BATCH * N_GT);
        const float cdF2 = sRowF2 / (float)(BATCH * N_FINE2)  + sColF2 / (float)(BATCH * N_GT);
        const float part = sPart  / (float)(BATCH * N_PART);
        const float sm   = sSm    / (float)(BATCH * N_FINE2 * KNN);
        const float total = 1.0f * cdC + 1.0f * cdF1 + 1.0f * cdF2 + 0.5f * part + 0.1f * sm;
        out[0] = total;
        out[1] = cdC;
        out[2] = cdF1;
        out[3] = cdF2;
        out[4] = part;
        out[5] = sm;
    }
}

// -------------------------------------------------------------------------
extern "C" void kernel_launch(void* const* d_in, const int* in_sizes, int n_in,
                              void* d_out, int out_size, void* d_ws, size_t ws_size,
                              hipStream_t stream) {
    const float* partial = (const float*)d_in[0];
    const float* coarse  = (const float*)d_in[1];
    const float* fine1   = (const float*)d_in[2];
    const float* fine2   = (const float*)d_in[3];
    const float* gt      = (const float*)d_in[4];
    float* ws  = (float*)d_ws;
    float* out = (float*)d_out;
    (void)in_sizes; (void)n_in; (void)out_size; (void)ws_size;

    init_colmin_kernel<<<dim3((3 * BATCH * N_GT + 255) / 256), 256, 0, stream>>>(ws);

    pairdist_min_kernel<true><<<dim3(N_COARSE / (16 * RTILES), BATCH), 32, 0, stream>>>(
        coarse, gt, ws + OFF_ROW_C, (unsigned int*)(ws + OFF_COL_C), N_COARSE, N_GT);
    pairdist_min_kernel<true><<<dim3(N_FINE1 / (16 * RTILES), BATCH), 32, 0, stream>>>(
        fine1, gt, ws + OFF_ROW_F1, (unsigned int*)(ws + OFF_COL_F1), N_FINE1, N_GT);
    pairdist_min_kernel<true><<<dim3(N_FINE2 / (16 * RTILES), BATCH), 32, 0, stream>>>(
        fine2, gt, ws + OFF_ROW_F2, (unsigned int*)(ws + OFF_COL_F2), N_FINE2, N_GT);
    // partial matching: one-directional (rows = partial, cols = fine2)
    pairdist_min_kernel<false><<<dim3(N_PART / (16 * RTILES), BATCH), 32, 0, stream>>>(
        partial, fine2, ws + OFF_ROW_P, nullptr, N_PART, N_FINE2);

    smooth_kernel<<<dim3(N_FINE2 / 128, BATCH), 128, 0, stream>>>(fine2, ws + OFF_SMOOTH);

    reduce_kernel<<<1, 256, 0, stream>>>(ws, out);
}